// SPDReLU_7576322310231
// MI455X (gfx1250) — compile-verified
//
#include <hip/hip_runtime.h>

// SPDReLU: batched 16x16 symmetric eigh via parallel (tournament) Jacobi,
// with all O(N^3) work (rotation application + reconstruction) done on the
// CDNA5 matrix pipe using V_WMMA_F32_16X16X4_F32 (16x16x16 f32 matmul = 4 WMMAs).
// One matrix per wave32; 8 waves (8 matrices) per 256-thread block.
//
// Synchronization note: each wave owns a PRIVATE LDS region, and CDNA5 LDS
// instructions from one wave complete in order (ISA 7.3), with all 32 lanes
// sharing a single instruction stream. So cross-lane LDS handoffs within a
// wave need only a *compiler* ordering fence, not s_barrier. WAVE_SYNC() is a
// zero-instruction memory clobber; no workgroup barriers are used at all,
// letting the 8 waves slide independently to hide LDS/WMMA latency.

typedef __attribute__((ext_vector_type(2))) float v2f;
typedef __attribute__((ext_vector_type(8))) float v8f;

#define SPD_SWEEPS 8          // 8 full cyclic sweeps (15 rounds each)
#define SPD_THRESH 1e-4f      // eigenvalue clamp from reference
#define SPD_RS     17         // LDS row stride (16+1) -> bank-conflict free
#define SPD_WAVE_LDS 864      // 3*16*17 + 32 (A,T,V + c/s arrays), padded

#define WAVE_SYNC() asm volatile("" ::: "memory")

// Round-robin (circle method) partner of index m in round rr (0..14).
// Index 15 is the fixed player; others rotate.
__device__ __forceinline__ int spd_partner(int m, int rr) {
    if (m == 15) return rr;
    if (m == rr) return 15;
    return (2 * rr + 15 - m) % 15;   // in [0,14], != m (since m != rr)
}

extern "C" __global__ void __launch_bounds__(256)
spdrelu_jacobi_wmma(const float* __restrict__ P, float* __restrict__ Out, int nmat)
{
    __shared__ float lds[8 * SPD_WAVE_LDS];

    const int tid  = threadIdx.x;
    const int wid  = tid >> 5;        // wave in block (0..7)
    const int lane = tid & 31;
    const int m    = lane & 15;       // A-layout row M / B-layout col N / C-layout col
    const int hi   = lane >> 4;       // half-wave select

    float* As = &lds[wid * SPD_WAVE_LDS];   // current (near-)diagonal A, row-major, stride 17
    float* Ts = As + 16 * SPD_RS;           // scratch T = A*J
    float* Vs = Ts + 16 * SPD_RS;           // accumulated eigenvector matrix V
    float* Cs = Vs + 16 * SPD_RS;           // per-index cos
    float* Ss = Cs + 16;                    // per-index signed sin (J[k][partner(k)])

    const int  b     = blockIdx.x * 8 + wid;
    const bool valid = (b < nmat);
    const int  bc    = valid ? b : (nmat - 1);       // clamp so all waves stay in-bounds
    const long long base = (long long)bc * 256;

    // ---- load A (C-layout row pattern: row = v + 8*hi, col = m), init V = I ----
#pragma unroll
    for (int v = 0; v < 8; ++v) {
        const int row = v + 8 * hi;
        As[row * SPD_RS + m] = P[base + row * 16 + m];
        Vs[row * SPD_RS + m] = (row == m) ? 1.0f : 0.0f;
    }
    WAVE_SYNC();

    // ---- parallel Jacobi: 8 disjoint Givens rotations per round ----
    for (int it = 0; it < SPD_SWEEPS * 15; ++it) {
        const int rr = it % 15;

        // lanes 0..7: compute rotation (c,s) for pair i, scatter per-index values
        if (lane < 8) {
            const int aI = (lane == 0) ? 15 : (rr + lane) % 15;
            const int bI = (lane == 0) ? rr : (rr - lane + 15) % 15;
            const int p  = aI < bI ? aI : bI;
            const int q  = aI < bI ? bI : aI;
            const float app = As[p * SPD_RS + p];
            const float aqq = As[q * SPD_RS + q];
            const float apq = As[p * SPD_RS + q];
            float c = 1.0f, s = 0.0f;
            if (__builtin_fabsf(apq) > 1e-20f) {
                const float theta = (aqq - app) / (2.0f * apq);
                const float t = __builtin_copysignf(1.0f, theta) /
                                (__builtin_fabsf(theta) + __builtin_sqrtf(theta * theta + 1.0f));
                c = 1.0f / __builtin_sqrtf(t * t + 1.0f);
                s = t * c;
            }
            // J = I except J[p][p]=J[q][q]=c, J[p][q]=s, J[q][p]=-s
            Cs[p] = c;  Cs[q] = c;
            Ss[p] = s;  Ss[q] = -s;
        }
        WAVE_SYNC();

        // build J in B-operand layout (k = 4*sl + v + 2*hi, n = m) and read A in A-layout
        v2f jB[4], aA[4];
#pragma unroll
        for (int sl = 0; sl < 4; ++sl) {
#pragma unroll
            for (int v = 0; v < 2; ++v) {
                const int k  = 4 * sl + v + 2 * hi;
                const int pk = spd_partner(k, rr);
                const float ck = Cs[k];
                const float sk = Ss[k];
                jB[sl][v] = (m == k) ? ck : ((m == pk) ? sk : 0.0f);
                aA[sl][v] = As[m * SPD_RS + k];          // A[m][k]
            }
        }

        // T = A * J  (16x16x16 f32 = 4 chained WMMAs, C-layout result)
        v8f t = {0.f,0.f,0.f,0.f,0.f,0.f,0.f,0.f};
#pragma unroll
        for (int sl = 0; sl < 4; ++sl)
            t = __builtin_amdgcn_wmma_f32_16x16x4_f32(false, aA[sl], false, jB[sl], 0, t, false, false);

        // spill T, then read T^T in A-layout (A symmetric => J^T A J = T^T J, reuse jB)
#pragma unroll
        for (int v = 0; v < 8; ++v)
            Ts[(v + 8 * hi) * SPD_RS + m] = t[v];
        WAVE_SYNC();

        v2f ttA[4], vA[4];
#pragma unroll
        for (int sl = 0; sl < 4; ++sl) {
#pragma unroll
            for (int v = 0; v < 2; ++v) {
                const int k = 4 * sl + v + 2 * hi;
                ttA[sl][v] = Ts[k * SPD_RS + m];         // (T^T)[m][k] = T[k][m]
                vA[sl][v]  = Vs[m * SPD_RS + k];         // V[m][k]
            }
        }

        v8f an = {0.f,0.f,0.f,0.f,0.f,0.f,0.f,0.f};
        v8f vn = {0.f,0.f,0.f,0.f,0.f,0.f,0.f,0.f};
#pragma unroll
        for (int sl = 0; sl < 4; ++sl) {
            an = __builtin_amdgcn_wmma_f32_16x16x4_f32(false, ttA[sl], false, jB[sl], 0, an, false, false);
            vn = __builtin_amdgcn_wmma_f32_16x16x4_f32(false, vA[sl],  false, jB[sl], 0, vn, false, false);
        }

        WAVE_SYNC();   // all reads of old A/V done before overwrite
#pragma unroll
        for (int v = 0; v < 8; ++v) {
            const int row = v + 8 * hi;
            As[row * SPD_RS + m] = an[v];
            Vs[row * SPD_RS + m] = vn[v];
        }
        WAVE_SYNC();
    }

    // ---- reconstruct: out = V * diag(max(w, 1e-4)) * V^T ----
    // M = V*diag(w) in A-layout; V^T in B-layout.  For this lane both read V[m][k].
    v2f mA[4], vtB[4];
#pragma unroll
    for (int sl = 0; sl < 4; ++sl) {
#pragma unroll
        for (int v = 0; v < 2; ++v) {
            const int k  = 4 * sl + v + 2 * hi;
            const float wk = __builtin_fmaxf(As[k * SPD_RS + k], SPD_THRESH);
            const float vk = Vs[m * SPD_RS + k];
            vtB[sl][v] = vk;          // V^T[k][n=m] = V[m][k]
            mA[sl][v]  = vk * wk;     // M[m][k]
        }
    }

    v8f o = {0.f,0.f,0.f,0.f,0.f,0.f,0.f,0.f};
#pragma unroll
    for (int sl = 0; sl < 4; ++sl)
        o = __builtin_amdgcn_wmma_f32_16x16x4_f32(false, mA[sl], false, vtB[sl], 0, o, false, false);

    if (valid) {
#pragma unroll
        for (int v = 0; v < 8; ++v)
            Out[base + (v + 8 * hi) * 16 + m] = o[v];   // coalesced: two 64B segments/row
    }
}

extern "C" void kernel_launch(void* const* d_in, const int* in_sizes, int n_in,
                              void* d_out, int out_size, void* d_ws, size_t ws_size,
                              hipStream_t stream)
{
    (void)n_in; (void)out_size; (void)d_ws; (void)ws_size;
    const float* P  = (const float*)d_in[0];
    float* Out      = (float*)d_out;
    const int nmat  = in_sizes[0] / 256;           // B = 262144 matrices of 16x16
    const int blocks = (nmat + 7) / 8;             // 8 matrices (waves) per block
    hipLaunchKernelGGL(spdrelu_jacobi_wmma, dim3(blocks), dim3(256), 0, stream,
                       P, Out, nmat);
}